// basic_gcn_50113678409885
// MI455X (gfx1250) — compile-verified
//
#include <hip/hip_runtime.h>
#include <hip/hip_bf16.h>

#define DFEAT 128
#define LDSP  132   // padded LDS row stride (floats) -> conflict-free ds reads, keeps 8/16B alignment
#define MT    64    // rows per block (4 x 16-row WMMA tiles per wave)

typedef float v2f __attribute__((ext_vector_type(2)));
typedef float v8f __attribute__((ext_vector_type(8)));

// ---------------- degree / norm precompute ----------------

__global__ void deg_init_k(float* __restrict__ deg, int n) {
    int i = blockIdx.x * blockDim.x + threadIdx.x;
    if (i < n) deg[i] = 1.0f;                 // self-loop weight
}

__global__ void deg_acc_k(const int* __restrict__ ei, const float* __restrict__ w,
                          float* __restrict__ deg, int E) {
    int e = blockIdx.x * blockDim.x + threadIdx.x;
    if (e < E) atomicAdd(deg + ei[E + e], w[e]);   // dst row of edge_index
}

__global__ void dinv_k(float* __restrict__ deg, int n) {
    int i = blockIdx.x * blockDim.x + threadIdx.x;
    if (i < n) {
        float d = deg[i];
        deg[i] = (d > 0.0f) ? rsqrtf(d) : 0.0f;    // in place: deg -> dinv
    }
}

__global__ void norm_k(const int* __restrict__ ei, const float* __restrict__ w,
                       const float* __restrict__ dinv, float* __restrict__ nrm, int E) {
    int e = blockIdx.x * blockDim.x + threadIdx.x;
    if (e < E) nrm[e] = dinv[ei[e]] * w[e] * dinv[ei[E + e]];
}

// ---------------- WMMA GEMM: out[n x 128] = (relu?)A[n x 128] @ W[128 x 128] (+ bias) ----------------
// One block = 256 threads = 8 waves. Block owns 64 rows; wave w owns output cols [16w, 16w+16)
// across FOUR 16-row tiles (4 accumulators). Each K-step loads one B fragment and feeds it to
// 4 x v_wmma_f32_16x16x4_f32 -> 4x B reuse + deep WMMA ILP.

__global__ __launch_bounds__(256)
void gemm_wmma_k(const float* __restrict__ A, const float* __restrict__ W,
                 const float* __restrict__ bias, float* __restrict__ out,
                 int nrows, int reluIn, int addBias) {
    __shared__ float As[MT][LDSP];    // 64 x 132 floats = ~33 KB

    const int tid = threadIdx.x;
    const int rowBase = blockIdx.x * MT;

    // Stage 64x128 A tile into LDS (float4 granularity; fold previous layer's ReLU here).
    const float4* A4 = (const float4*)A;
    #pragma unroll
    for (int j = 0; j < 8; ++j) {
        int idx4 = j * 256 + tid;        // 0..2047 float4 slots = 64 rows x 32 float4
        int r    = idx4 >> 5;
        int c4   = idx4 & 31;
        float4 v = make_float4(0.f, 0.f, 0.f, 0.f);
        int gr = rowBase + r;
        if (gr < nrows) v = A4[(size_t)gr * (DFEAT / 4) + c4];
        if (reluIn) {
            v.x = fmaxf(v.x, 0.f); v.y = fmaxf(v.y, 0.f);
            v.z = fmaxf(v.z, 0.f); v.w = fmaxf(v.w, 0.f);
        }
        *(float4*)&As[r][c4 * 4] = v;
    }
    __syncthreads();

    const int lane = tid & 31;
    const int wv   = tid >> 5;           // 0..7 column tile
    const int colBase = wv * 16;
    const int m  = lane & 15;            // M index (A) / N index (B,D)
    const int kh = (lane >> 4) * 2;      // lanes 16-31 carry K+2,K+3
    const int col = colBase + m;

    v8f acc[4] = {};
    #pragma unroll
    for (int k0 = 0; k0 < DFEAT; k0 += 4) {
        v2f b;                            // one B fragment per K-step, reused by 4 WMMAs
        b.x = W[(size_t)(k0 + kh)     * DFEAT + col];
        b.y = W[(size_t)(k0 + kh + 1) * DFEAT + col];
        #pragma unroll
        for (int rt = 0; rt < 4; ++rt) {
            v2f a;
            a.x = As[rt * 16 + m][k0 + kh];
            a.y = As[rt * 16 + m][k0 + kh + 1];
            acc[rt] = __builtin_amdgcn_wmma_f32_16x16x4_f32(
                          false, a, false, b, (short)0, acc[rt], false, false);
        }
    }

    float bv = 0.0f;
    if (addBias) bv = bias[col];
    const int rowOff = (lane >> 4) * 8;  // D layout: lanes>=16 hold M=8..15
    #pragma unroll
    for (int rt = 0; rt < 4; ++rt) {
        #pragma unroll
        for (int v = 0; v < 8; ++v) {
            int gr = rowBase + rt * 16 + rowOff + v;
            if (gr < nrows) out[(size_t)gr * DFEAT + col] = acc[rt][v] + bv;
        }
    }
}

// ---------------- aggregation: out = h * dinv^2 (self loop) + bias, then scatter edges ----------------

__global__ void agg_init_k(const float* __restrict__ hb, const float* __restrict__ dinv,
                           const float* __restrict__ bias, float* __restrict__ out, int n) {
    int gid = blockIdx.x * blockDim.x + threadIdx.x;   // n*32 float4 slots
    if (gid >= n * (DFEAT / 4)) return;
    int row = gid >> 5;
    int c4  = gid & 31;
    float s = dinv[row];
    s = s * s;                                          // dinv[i]*1*dinv[i]
    float4 v  = ((const float4*)hb)[gid];
    float4 bv = ((const float4*)bias)[c4];
    float4 r;
    r.x = v.x * s + bv.x; r.y = v.y * s + bv.y;
    r.z = v.z * s + bv.z; r.w = v.w * s + bv.w;
    ((float4*)out)[gid] = r;
}

// One wave per edge; lane moves float4 (coalesced 512B gather + 4 L2-resident f32 atomics).
__global__ __launch_bounds__(256)
void scatter_k(const float* __restrict__ hb, const int* __restrict__ ei,
               const float* __restrict__ nrm, float* __restrict__ out, int E) {
    int gid  = blockIdx.x * blockDim.x + threadIdx.x;
    int e    = gid >> 5;
    int lane = gid & 31;
    if (e >= E) return;
    int s = ei[e];
    int d = ei[E + e];
    float nv = nrm[e];
    float4 v = ((const float4*)(hb + (size_t)s * DFEAT))[lane];
    float* o = out + (size_t)d * DFEAT + lane * 4;
    atomicAdd(o + 0, v.x * nv);
    atomicAdd(o + 1, v.y * nv);
    atomicAdd(o + 2, v.z * nv);
    atomicAdd(o + 3, v.w * nv);
}

// ---------------- launch ----------------

static inline size_t align_up(size_t x) { return (x + 63) & ~(size_t)63; }

extern "C" void kernel_launch(void* const* d_in, const int* in_sizes, int n_in,
                              void* d_out, int out_size, void* d_ws, size_t ws_size,
                              hipStream_t stream) {
    const float* x  = (const float*)d_in[0];
    const int*   ei = (const int*)d_in[1];
    const float* ew = (const float*)d_in[2];
    const float* W1 = (const float*)d_in[3];  const float* b1 = (const float*)d_in[4];
    const float* W2 = (const float*)d_in[5];  const float* b2 = (const float*)d_in[6];
    const float* W3 = (const float*)d_in[7];  const float* b3 = (const float*)d_in[8];
    const float* Wl = (const float*)d_in[9];  const float* bl = (const float*)d_in[10];

    const int N = in_sizes[0] / DFEAT;
    const int E = in_sizes[2];

    // workspace carve-up (floats): dinv[N] | norm[E] | hb[N*D] | bufA[N*D] | bufB[N*D]
    float* dinv = (float*)d_ws;
    float* nrm  = dinv + align_up((size_t)N);
    float* hb   = nrm  + align_up((size_t)E);
    float* bufA = hb   + (size_t)N * DFEAT;
    float* bufB = bufA + (size_t)N * DFEAT;
    float* outF = (float*)d_out;

    const int T = 256;
    dim3 blkN((N + T - 1) / T), blkE((E + T - 1) / T);
    dim3 blkG((N + MT - 1) / MT);                   // GEMM: 64 rows per block
    dim3 blkI(((size_t)N * (DFEAT / 4) + T - 1) / T);
    dim3 blkS(((size_t)E * 32 + T - 1) / T);        // 1 wave per edge

    // norm precompute
    deg_init_k<<<blkN, T, 0, stream>>>(dinv, N);
    deg_acc_k <<<blkE, T, 0, stream>>>(ei, ew, dinv, E);
    dinv_k    <<<blkN, T, 0, stream>>>(dinv, N);
    norm_k    <<<blkE, T, 0, stream>>>(ei, ew, dinv, nrm, E);

    // layer 1: h = x @ W1 ; agg ; (relu folded into next gemm's A load)
    gemm_wmma_k<<<blkG, T, 0, stream>>>(x,    W1, b1, hb,   N, 0, 0);
    agg_init_k <<<blkI, T, 0, stream>>>(hb, dinv, b1, bufA, N);
    scatter_k  <<<blkS, T, 0, stream>>>(hb, ei, nrm, bufA, E);

    // layer 2
    gemm_wmma_k<<<blkG, T, 0, stream>>>(bufA, W2, b2, hb,   N, 1, 0);
    agg_init_k <<<blkI, T, 0, stream>>>(hb, dinv, b2, bufB, N);
    scatter_k  <<<blkS, T, 0, stream>>>(hb, ei, nrm, bufB, E);

    // layer 3 (no relu after)
    gemm_wmma_k<<<blkG, T, 0, stream>>>(bufB, W3, b3, hb,   N, 1, 0);
    agg_init_k <<<blkI, T, 0, stream>>>(hb, dinv, b3, bufA, N);
    scatter_k  <<<blkS, T, 0, stream>>>(hb, ei, nrm, bufA, E);

    // final linear: d_out = h3 @ Wl + bl
    gemm_wmma_k<<<blkG, T, 0, stream>>>(bufA, Wl, bl, outF, N, 0, 1);
}